// Evaluator_37632503448234
// MI455X (gfx1250) — compile-verified
//
#include <hip/hip_runtime.h>

// Problem constants (reference: target [5, 2048, 2048] float32).
#define H    2048
#define W    2048
#define HW   (H * W)
#define NCH  5

// Stencil constants: kernel 9x9, radius 4, center skipped.
#define HALO 4
#define TILE 32
#define LW   (TILE + 2 * HALO)   // 40
#define LSZ  (LW * LW)           // 1600 bytes of u8 tile

// Padded combined-mask plane (replicate pad baked in by pass 1).
#define PW   (W + 2 * HALO)      // 2056
#define PH   (H + 2 * HALO)      // 2056
#define PHW  (PH * PW)

// Workspace layout (d_ws):
//   [0, HW*4)            : feature_weight, float          (16 MB)
//   [HW*4, HW*4+PHW)     : padded combined mask, u8       (~4.2 MB)
//   then (8B aligned)    : {min_bits, max_bits} u32

#define WS_RED_OFF ((((size_t)HW * 4 + (size_t)PHW) + 7) & ~(size_t)7)

// Low 32 bits of a flat pointer into the LDS aperture == LDS byte offset.
__device__ __forceinline__ unsigned lds_off(const void* p) {
  return (unsigned)(unsigned long long)p;
}

#if __has_builtin(__builtin_amdgcn_tensor_load_to_lds)
#define USE_TDM 1
typedef unsigned int v4u __attribute__((ext_vector_type(4)));
typedef int v4i __attribute__((ext_vector_type(4)));
typedef int v8i __attribute__((ext_vector_type(8)));

// One TDM descriptor: copy a LW x LW u8 tile (row stride PW bytes) from
// global memory into LDS. D# per CDNA5 ISA ch.8.3/8.4:
//   group0: count=1 | lds_addr | global_addr[56:0] | type=2
//   group1: data_size=1B, tensor_dim0/1 = tile_dim0/1 = LW (no OOB),
//           tensor_dim0_stride = PW
__device__ __forceinline__ void tdm_load_tile(unsigned lds_addr,
                                              const void* gtile) {
  const unsigned long long ga = (unsigned long long)gtile;
  v4u g0;
  g0.x = 1u;                                        // count=1, user mode
  g0.y = lds_addr;                                  // LDS byte address
  g0.z = (unsigned)(ga & 0xFFFFFFFFu);              // global_addr[31:0]
  g0.w = (unsigned)((ga >> 32) & 0x01FFFFFFu)       // global_addr[56:32]
         | (2u << 30);                              // type=2 ("image")

  v8i g1;
  g1[0] = 0;                        // wg_mask=0, data_size=0(1B), no flags
  g1[1] = (int)((unsigned)LW << 16);        // tensor_dim0[15:0] (=LW)
  g1[2] = (int)((unsigned)LW << 16);        // dim0[31:16]=0 | tensor_dim1[15:0]
  g1[3] = (int)((unsigned)LW << 16);        // dim1[31:16]=0 | tile_dim0=LW
  g1[4] = (int)(unsigned)LW;                // tile_dim1=LW | tile_dim2=0
  g1[5] = (int)(unsigned)PW;                // tensor_dim0_stride[31:0]
  g1[6] = 0;                                // stride0[47:32] | stride1[15:0]
  g1[7] = 0;                                // stride1[47:16]

  const v4i z4 = {0, 0, 0, 0};              // groups 2/3 unused (2D tile)
#if __clang_major__ >= 23
  const v8i z8 = {0, 0, 0, 0, 0, 0, 0, 0};
  __builtin_amdgcn_tensor_load_to_lds(g0, g1, z4, z4, z8, 0);
#else
  __builtin_amdgcn_tensor_load_to_lds(g0, g1, z4, z4, 0);
#endif
}
#else
// Fallback: per-lane async global->LDS byte copies (ASYNCcnt-tracked).
__device__ __forceinline__ void async_ld_b8(unsigned lds_byte_addr,
                                            unsigned long long gaddr) {
  asm volatile("global_load_async_to_lds_b8 %0, %1, off"
               :
               : "v"(lds_byte_addr), "v"(gaddr)
               : "memory");
}
__device__ __forceinline__ void wait_async0() {
  asm volatile("s_wait_asynccnt 0" ::: "memory");
}
#endif

// ---- Pass 1: clamped channel sum -> fw (f32) + PADDED combined (u8) -------
__global__ __launch_bounds__(256) void pass1_kernel(
    const float* __restrict__ target, float* __restrict__ fw,
    unsigned char* __restrict__ combp, unsigned* __restrict__ red) {
  const int i = blockIdx.x * 256 + threadIdx.x;
  if (blockIdx.x == 0 && threadIdx.x == 0) {
    red[0] = 0x7F800000u;  // +inf  (running min of w >= 0)
    red[1] = 0u;           // 0.0f  (running max of w >= 0)
  }
  if (i >= PHW) return;
  const int py = i / PW;
  const int px = i - py * PW;
  int y = py - HALO;
  y = y < 0 ? 0 : (y > H - 1 ? H - 1 : y);
  int x = px - HALO;
  x = x < 0 ? 0 : (x > W - 1 ? W - 1 : x);
  const size_t g = (size_t)y * W + x;
  const float s = target[g] + target[g + HW] + target[g + 2 * (size_t)HW] +
                  target[g + 3 * (size_t)HW] + target[g + 4 * (size_t)HW];
  combp[i] = (s > 0.0f) ? (unsigned char)1 : (unsigned char)0;
  if ((unsigned)(py - HALO) < (unsigned)H &&
      (unsigned)(px - HALO) < (unsigned)W) {
    fw[(size_t)(py - HALO) * W + (px - HALO)] = s;
  }
}

// ---- Pass 2: 80-tap min-distance stencil out of a TDM-staged LDS tile -----
__global__ __launch_bounds__(256) void pass2_kernel(
    const float* __restrict__ fw, const unsigned char* __restrict__ combp,
    float* __restrict__ out, unsigned* __restrict__ red) {
  __shared__ unsigned char sm[LSZ];
  __shared__ unsigned s_min, s_max;

  const int tid = threadIdx.y * 32 + threadIdx.x;  // blockDim = (32, 8)
  if (tid == 0) {
    s_min = 0x7F800000u;
    s_max = 0u;
  }

  const int bx = blockIdx.x * TILE;
  const int by = blockIdx.y * TILE;

#ifdef USE_TDM
  // One TDM descriptor per workgroup; TDM ignores EXEC, so gate with a
  // scalar (SGPR) branch: only wave 0 issues + waits, barrier publishes.
  if (__builtin_amdgcn_readfirstlane(tid) == 0) {
    tdm_load_tile(lds_off(&sm[0]),
                  (const void*)(combp + (size_t)by * PW + bx));
    __builtin_amdgcn_s_wait_tensorcnt(0);
  }
#else
  for (int idx = tid; idx < LSZ; idx += 256) {
    const int ly = idx / LW;
    const int lx = idx - ly * LW;
    async_ld_b8(lds_off(&sm[idx]),
                (unsigned long long)(const void*)(combp +
                                                  (size_t)(by + ly) * PW +
                                                  (bx + lx)));
  }
  wait_async0();
#endif
  __syncthreads();  // tile + s_min/s_max init visible to all waves

  const float INF = __uint_as_float(0x7F800000u);
  float lmin = INF, lmax = 0.0f;

  const int tx = threadIdx.x;
#pragma unroll
  for (int k = 0; k < 4; ++k) {
    const int ty = threadIdx.y + k * 8;  // rows 0..31 of the tile
    const unsigned char c = sm[(ty + HALO) * LW + (tx + HALO)];

    float mind = INF;
#pragma unroll
    for (int dy = -HALO; dy <= HALO; ++dy) {
#pragma unroll
      for (int dx = -HALO; dx <= HALO; ++dx) {
        if (dy == 0 && dx == 0) continue;
        const unsigned char n = sm[(ty + dy + HALO) * LW + (tx + dx + HALO)];
        // sqrtf of a compile-time constant: folded after full unroll.
        const float kd = sqrtf((float)(dy * dy + dx * dx));
        if (n != c) mind = fminf(mind, kd);
      }
    }
    // fp32: (diff + 1e-10) == diff and (min_d + 1e-10) == min_d exactly.
    const float contour = (mind < INF) ? (1.0f / mind) : 0.0f;
    const size_t g = (size_t)(by + ty) * W + (bx + tx);
    float wv = fw[g] + contour;
    wv *= wv;
    out[g] = wv;
    lmin = fminf(lmin, wv);
    lmax = fmaxf(lmax, wv);
  }

  // Non-negative floats: uint bit pattern is order-preserving.
  atomicMin(&s_min, __float_as_uint(lmin));
  atomicMax(&s_max, __float_as_uint(lmax));
  __syncthreads();
  if (tid == 0) {
    atomicMin(&red[0], s_min);
    atomicMax(&red[1], s_max);
  }
}

// ---- Pass 3: global min-max normalize in place, mask by combined ----------
__global__ __launch_bounds__(256) void pass3_kernel(
    float* __restrict__ out, const unsigned char* __restrict__ combp,
    const unsigned* __restrict__ red) {
  const int i = blockIdx.x * 256 + threadIdx.x;
  if (i >= HW) return;
  const int y = i / W;
  const int x = i - y * W;
  const float mn = __uint_as_float(red[0]);
  const float mx = __uint_as_float(red[1]);
  const float wv = out[i];
  const float c = (float)combp[(size_t)(y + HALO) * PW + (x + HALO)];
  out[i] = (wv - mn) / (mx - mn + 1e-10f) * c;
}

extern "C" void kernel_launch(void* const* d_in, const int* in_sizes, int n_in,
                              void* d_out, int out_size, void* d_ws,
                              size_t ws_size, hipStream_t stream) {
  (void)in_sizes; (void)n_in; (void)out_size; (void)ws_size;
  const float* target = (const float*)d_in[0];
  float* out = (float*)d_out;

  char* ws = (char*)d_ws;
  float* fw = (float*)ws;
  unsigned char* combp = (unsigned char*)(ws + (size_t)HW * 4);
  unsigned* red = (unsigned*)(ws + WS_RED_OFF);

  pass1_kernel<<<(PHW + 255) / 256, 256, 0, stream>>>(target, fw, combp, red);

  dim3 grid(W / TILE, H / TILE);
  dim3 blk(32, 8);
  pass2_kernel<<<grid, blk, 0, stream>>>(fw, combp, out, red);

  pass3_kernel<<<(HW + 255) / 256, 256, 0, stream>>>(out, combp, red);
}